// ForwardWarp_36928128811543
// MI455X (gfx1250) — compile-verified
//
#include <hip/hip_runtime.h>

// Problem constants (reference: N,C,H,W = 8,3,1024,1024)
#define N_  8
#define C_  3
#define H_  1024
#define W_  1024
#define HW_ (H_ * W_)

typedef __attribute__((ext_vector_type(2))) float v2f;

// Non-returning fp32 atomic add, relaxed, device scope.
// Lowers to GLOBAL_ATOMIC_ADD_F32 (no return -> TH[0]=0, no wait needed).
__device__ __forceinline__ void atom_add_f32(float* p, float v) {
    __hip_atomic_fetch_add(p, v, __ATOMIC_RELAXED, __HIP_MEMORY_SCOPE_AGENT);
}

// One thread handles 2 adjacent pixels (w, w+1) of one (n,h) row.
// Streaming inputs use non-temporal 64-bit loads so the 167MB of inputs do
// not evict the ~200MB of output accumulators from the 192MB global L2;
// the 4-corner gaussian splat is done with hardware f32 atomics that
// resolve in L2.
__global__ __launch_bounds__(256) void fw_splat_kernel(const float* __restrict__ img,
                                                       const float* __restrict__ flo,
                                                       float* __restrict__ imgw,
                                                       float* __restrict__ onew) {
    int t   = blockIdx.x * 256 + threadIdx.x;   // 0 .. N*H*W/2 - 1
    int pix = t << 1;                           // even pixel index
    int hw  = pix & (HW_ - 1);
    int n   = pix >> 20;                        // HW_ == 2^20
    int h   = hw >> 10;                         // row
    int w   = hw & (W_ - 1);                    // even col

    // Reference convention: flo[:,0] = column (W) shift, flo[:,1] = row (H) shift.
    const float* flo_n = flo + (size_t)n * 2 * HW_;
    v2f yf2 = __builtin_nontemporal_load((const v2f*)(flo_n + hw));        // col shift
    v2f xf2 = __builtin_nontemporal_load((const v2f*)(flo_n + HW_ + hw));  // row shift

    const float* img_n = img + (size_t)n * C_ * HW_;
    v2f i0v = __builtin_nontemporal_load((const v2f*)(img_n + hw));
    v2f i1v = __builtin_nontemporal_load((const v2f*)(img_n + HW_ + hw));
    v2f i2v = __builtin_nontemporal_load((const v2f*)(img_n + 2 * HW_ + hw));

    int base = n * C_ * HW_;   // channel-0 base for this image

#pragma unroll
    for (int p = 0; p < 2; ++p) {
        float yf = yf2[p];
        float xf = xf2[p];
        float i0 = i0v[p];
        float i1 = i1v[p];
        float i2 = i2v[p];
        int   wp = w + p;

        float x1 = floorf(xf);
        float y1 = floorf(yf);
        int ix0 = (int)x1 + h;     // destination row for dx=0
        int iy0 = (int)y1 + wp;    // destination col for dy=0

        // Fractional offsets; distance for the +1 corner is (frac - 1).
        float fx = xf - x1;
        float fy = yf - y1;
        float ex0 = fx * fx;
        float ex1 = (fx - 1.f) * (fx - 1.f);
        float ey0 = fy * fy;
        float ey1 = (fy - 1.f) * (fy - 1.f);

#pragma unroll
        for (int dx = 0; dx < 2; ++dx) {
            int ix = ix0 + dx;
            if (ix < 0 || ix >= H_) continue;
            float ex = dx ? ex1 : ex0;
#pragma unroll
            for (int dy = 0; dy < 2; ++dy) {
                int iy = iy0 + dy;
                if (iy < 0 || iy >= W_) continue;
                float ey  = dy ? ey1 : ey0;
                float wgt = __expf(-(ex + ey));
                int id = base + ix * W_ + iy;
                // 3 channels x (imgw, onew); weight identical across channels.
                atom_add_f32(imgw + id,           i0 * wgt);
                atom_add_f32(imgw + id + HW_,     i1 * wgt);
                atom_add_f32(imgw + id + 2 * HW_, i2 * wgt);
                atom_add_f32(onew + id,           wgt);
                atom_add_f32(onew + id + HW_,     wgt);
                atom_add_f32(onew + id + 2 * HW_, wgt);
            }
        }
    }
}

// Zero both output accumulators with regular-temporal stores so the
// accumulator lines are resident in the 192MB L2 when the splat kernel's
// atomics arrive.
__global__ __launch_bounds__(256) void fw_zero_kernel(float4* __restrict__ out, int n4) {
    int i = blockIdx.x * 256 + threadIdx.x;
    if (i < n4) out[i] = make_float4(0.f, 0.f, 0.f, 0.f);
}

extern "C" void kernel_launch(void* const* d_in, const int* in_sizes, int n_in,
                              void* d_out, int out_size, void* d_ws, size_t ws_size,
                              hipStream_t stream) {
    (void)in_sizes; (void)n_in; (void)d_ws; (void)ws_size;

    const float* img = (const float*)d_in[0];   // (8,3,1024,1024) f32
    const float* flo = (const float*)d_in[1];   // (8,2,1024,1024) f32

    float* imgw = (float*)d_out;                          // first output
    float* onew = imgw + (size_t)N_ * C_ * HW_;           // second output

    // Zero the concatenated outputs (50,331,648 floats, divisible by 4).
    int total = 2 * N_ * C_ * HW_;
    int n4 = total / 4;
    fw_zero_kernel<<<(n4 + 255) / 256, 256, 0, stream>>>((float4*)d_out, n4);

    // One thread per 2 source pixels: 4,194,304 threads.
    int nthreads = (N_ * HW_) / 2;
    fw_splat_kernel<<<nthreads / 256, 256, 0, stream>>>(img, flo, imgw, onew);
}